// GATPropagation_16286515986840
// MI455X (gfx1250) — compile-verified
//
#include <hip/hip_runtime.h>
#include <hip/hip_bf16.h>
#include <math.h>

typedef __attribute__((ext_vector_type(2))) float v2f;
typedef __attribute__((ext_vector_type(8))) float v8f;

#define NEG_SLOPE 0.2f

// order-preserving float max via integer atomics (native global atomics)
__device__ __forceinline__ void atomic_max_f32(float* addr, float val) {
  if (val >= 0.f) {
    atomicMax((int*)addr, __float_as_int(val));
  } else {
    atomicMin((unsigned int*)addr, __float_as_uint(val));
  }
}

__global__ void fill_kernel(float* __restrict__ p, float v, int n) {
  int i = blockIdx.x * blockDim.x + threadIdx.x;
  if (i < n) p[i] = v;
}

// p[i] = 1 / max(p[i], 1e-16)  (hoists the softmax divide out of the edge scatter)
__global__ void inv_kernel(float* __restrict__ p, int n) {
  int i = blockIdx.x * blockDim.x + threadIdx.x;
  if (i < n) p[i] = 1.f / fmaxf(p[i], 1e-16f);
}

// C[M x NCOLS] = A[M x K] @ B[K x NCOLS], all row-major fp32.
// grid.x = ceil(M/16); blockDim.x = 32 * (NCOLS/16) waves; wave w owns cols [16w,16w+16).
// A 16xK tile staged into LDS with async global->LDS B128 copies; fp32 WMMA 16x16x4.
template<int K, int NCOLS>
__global__ void gemm_wmma_kernel(const float* __restrict__ A,
                                 const float* __restrict__ B,
                                 float* __restrict__ C,
                                 int M) {
  __shared__ float Atile[16 * K];   // only __shared__ object -> LDS offset base 0
  const int rowbase = blockIdx.x * 16;
  const int tid = threadIdx.x;

  if (rowbase + 16 <= M) {
    // full tile: async DMA global -> LDS, 16 bytes per lane per issue
    for (int i = tid * 4; i < 16 * K; i += blockDim.x * 4) {
      const int r = i / K, c = i - r * K;
      const unsigned lds_off = (unsigned)i * 4u;          // byte offset in LDS
      const float* gp = A + (size_t)(rowbase + r) * K + c;
      asm volatile("global_load_async_to_lds_b128 %0, %1, off"
                   :: "v"(lds_off), "v"(gp) : "memory");
    }
    asm volatile("s_wait_asynccnt 0x0" ::: "memory");
  } else {
    for (int i = tid; i < 16 * K; i += blockDim.x) {
      const int r = i / K, c = i - r * K;
      const int gr = rowbase + r;
      Atile[i] = (gr < M) ? A[(size_t)gr * K + c] : 0.f;
    }
  }
  __syncthreads();

  const int lane = tid & 31;
  const int colbase = (tid >> 5) << 4;                    // wave-uniform
  const int half = lane >> 4;                             // 0 or 1
  const int l15 = lane & 15;
  v8f acc = {};
  const float* Bcol = B + colbase + l15;
  // A frag (16x4): v0 = K=0 (lanes 0-15) / K=2 (lanes 16-31); v1 = K=1 / K=3
  // B frag (4x16): mirrored K split, N striped across lanes
#pragma unroll 4
  for (int k0 = 0; k0 < K; k0 += 4) {
    const int kk = k0 + 2 * half;
    v2f a, b;
    a.x = Atile[l15 * K + kk];
    a.y = Atile[l15 * K + kk + 1];
    b.x = Bcol[kk * NCOLS];
    b.y = Bcol[(kk + 1) * NCOLS];
    acc = __builtin_amdgcn_wmma_f32_16x16x4_f32(false, a, false, b,
                                                (short)0, acc, false, false);
  }
  // C/D layout: VGPR v, lanes 0-15 -> M=v, lanes 16-31 -> M=v+8; N = lane&15
  float* Cp = C + ((size_t)rowbase + 8 * half) * NCOLS + colbase + l15;
  if (rowbase + 16 <= M) {
#pragma unroll
    for (int v = 0; v < 8; ++v) Cp[v * NCOLS] = acc[v];
  } else {
#pragma unroll
    for (int v = 0; v < 8; ++v)
      if (rowbase + 8 * half + v < M) Cp[v * NCOLS] = acc[v];
  }
}

// s[n,h] = sum_c h[n,h,c]*a_src[h,c] ; d[n,h] likewise. One thread per (node,head).
template<int H, int C>
__global__ void rowdot_kernel(const float* __restrict__ h,
                              const float* __restrict__ a_src,
                              const float* __restrict__ a_dst,
                              float* __restrict__ s_out, float* __restrict__ d_out_,
                              int n_nodes) {
  int idx = blockIdx.x * blockDim.x + threadIdx.x;
  if (idx >= n_nodes * H) return;
  const int head = idx & (H - 1);
  const float* hp = h + (size_t)idx * C;   // h is [n][H][C]; idx = n*H+head
  const float* as = a_src + head * C;
  const float* ad = a_dst + head * C;
  float ss = 0.f, dd = 0.f;
#pragma unroll
  for (int c = 0; c < C; ++c) {
    float v = hp[c];
    ss = fmaf(v, as[c], ss);
    dd = fmaf(v, ad[c], dd);
  }
  s_out[idx] = ss;
  d_out_[idx] = dd;
}

__device__ __forceinline__ void edge_nodes(const int* __restrict__ ei, int E, int e,
                                           int& sn, int& dn) {
  if (e < E) { sn = ei[e]; dn = ei[E + e]; }
  else       { sn = e - E; dn = sn; }          // self-loops appended
}

template<int H>
__global__ void edge_alpha_max_kernel(const int* __restrict__ ei, int E, int EE,
                                      const float* __restrict__ s,
                                      const float* __restrict__ d,
                                      float* __restrict__ mx) {
  int idx = blockIdx.x * blockDim.x + threadIdx.x;
  if (idx >= EE * H) return;
  const int e = idx / H, h = idx & (H - 1);
  int sn, dn; edge_nodes(ei, E, e, sn, dn);
  float a = s[sn * H + h] + d[dn * H + h];
  a = (a > 0.f) ? a : a * NEG_SLOPE;
  atomic_max_f32(&mx[dn * H + h], a);
}

template<int H>
__global__ void edge_exp_kernel(const int* __restrict__ ei, int E, int EE,
                                const float* __restrict__ s,
                                const float* __restrict__ d,
                                const float* __restrict__ mx,
                                float* __restrict__ dsum,
                                float* __restrict__ ew) {
  int idx = blockIdx.x * blockDim.x + threadIdx.x;
  if (idx >= EE * H) return;
  const int e = idx / H, h = idx & (H - 1);
  int sn, dn; edge_nodes(ei, E, e, sn, dn);
  float a = s[sn * H + h] + d[dn * H + h];
  a = (a > 0.f) ? a : a * NEG_SLOPE;
  float ex = __expf(a - mx[dn * H + h]);
  ew[idx] = ex;
  atomicAdd(&dsum[dn * H + h], ex);
}

// one wave32 per edge; lane handles HC/32 contiguous channels.
// dninv = precomputed 1/max(denom,1e-16).
template<int HC, int H>
__global__ void edge_scatter_kernel(const int* __restrict__ ei, int E, int EE,
                                    const float* __restrict__ feat,
                                    const float* __restrict__ ew,
                                    const float* __restrict__ dninv,
                                    float* __restrict__ outacc) {
  int g = blockIdx.x * blockDim.x + threadIdx.x;
  int e = g >> 5;
  int lane = g & 31;
  if (e >= EE) return;
  int sn, dn; edge_nodes(ei, E, e, sn, dn);
  constexpr int PER = HC / 32;
  constexpr int C = HC / H;
  const int c0 = lane * PER;
  const int head = c0 / C;                     // constant across PER channels
  const float w = ew[(size_t)e * H + head] * dninv[(size_t)dn * H + head];
  float* od = outacc + (size_t)dn * HC + c0;
  if constexpr (PER == 4) {
    const float4 f4 = *(const float4*)(feat + (size_t)sn * HC + c0);  // b128 gather
    atomicAdd(&od[0], f4.x * w);
    atomicAdd(&od[1], f4.y * w);
    atomicAdd(&od[2], f4.z * w);
    atomicAdd(&od[3], f4.w * w);
  } else {
    const float* fs = feat + (size_t)sn * HC + c0;
#pragma unroll
    for (int j = 0; j < PER; ++j) atomicAdd(&od[j], fs[j] * w);
  }
}

template<int F>
__global__ void bias_relu_kernel(float* __restrict__ x, const float* __restrict__ b,
                                 int total) {
  int i = blockIdx.x * blockDim.x + threadIdx.x;
  if (i >= total) return;
  x[i] = fmaxf(x[i] + b[i & (F - 1)], 0.f);
}

template<int F>
__global__ void bias_sigmoid_kernel(const float* __restrict__ x, const float* __restrict__ b,
                                    float* __restrict__ out, int total) {
  int i = blockIdx.x * blockDim.x + threadIdx.x;
  if (i >= total) return;
  float v = x[i] + b[i & (F - 1)];
  out[i] = 1.f / (1.f + __expf(-v));
}

static inline int cdiv(int a, int b) { return (a + b - 1) / b; }

extern "C" void kernel_launch(void* const* d_in, const int* in_sizes, int n_in,
                              void* d_out, int out_size, void* d_ws, size_t ws_size,
                              hipStream_t stream) {
  const float* x      = (const float*)d_in[0];
  const int*   ei     = (const int*)d_in[1];
  const float* W1     = (const float*)d_in[2];
  const float* a_src1 = (const float*)d_in[3];
  const float* a_dst1 = (const float*)d_in[4];
  const float* b1     = (const float*)d_in[5];
  const float* W2     = (const float*)d_in[6];
  const float* a_src2 = (const float*)d_in[7];
  const float* a_dst2 = (const float*)d_in[8];
  const float* b2     = (const float*)d_in[9];

  const int Nn = in_sizes[0] / 128;   // 50000
  const int E  = in_sizes[1] / 2;     // 800000
  const int EE = E + Nn;              // + self-loops

  float* ws   = (float*)d_ws;
  float* h1   = ws;                              // Nn*128
  float* out1 = h1 + (size_t)Nn * 128;           // Nn*128 (becomes x2 in place)
  float* s1   = out1 + (size_t)Nn * 128;         // Nn*4
  float* d1   = s1 + (size_t)Nn * 4;             // Nn*4
  float* mx1  = d1 + (size_t)Nn * 4;             // Nn*4
  float* dn1  = mx1 + (size_t)Nn * 4;            // Nn*4
  float* ew1  = dn1 + (size_t)Nn * 4;            // EE*4
  // layer-2 reuses layer-1 regions
  float* h2   = h1;                              // Nn*32
  float* out2 = h1 + (size_t)Nn * 32;            // Nn*32
  float *s2 = s1, *d2 = d1, *mx2 = mx1, *dn2 = dn1, *ew2 = ew1;

  const float NEGINF = -3.402823466e38f;
  dim3 b256(256);

  // ---------------- layer 1 ----------------
  gemm_wmma_kernel<128, 128><<<cdiv(Nn, 16), 256, 0, stream>>>(x, W1, h1, Nn);
  rowdot_kernel<4, 32><<<cdiv(Nn * 4, 256), b256, 0, stream>>>(h1, a_src1, a_dst1, s1, d1, Nn);
  fill_kernel<<<cdiv(Nn * 4, 256), b256, 0, stream>>>(mx1, NEGINF, Nn * 4);
  fill_kernel<<<cdiv(Nn * 4, 256), b256, 0, stream>>>(dn1, 0.f, Nn * 4);
  fill_kernel<<<cdiv(Nn * 128, 256), b256, 0, stream>>>(out1, 0.f, Nn * 128);
  edge_alpha_max_kernel<4><<<cdiv(EE * 4, 256), b256, 0, stream>>>(ei, E, EE, s1, d1, mx1);
  edge_exp_kernel<4><<<cdiv(EE * 4, 256), b256, 0, stream>>>(ei, E, EE, s1, d1, mx1, dn1, ew1);
  inv_kernel<<<cdiv(Nn * 4, 256), b256, 0, stream>>>(dn1, Nn * 4);
  edge_scatter_kernel<128, 4><<<cdiv(EE * 32, 256), b256, 0, stream>>>(ei, E, EE, h1, ew1, dn1, out1);
  bias_relu_kernel<128><<<cdiv(Nn * 128, 256), b256, 0, stream>>>(out1, b1, Nn * 128);

  // ---------------- layer 2 ----------------
  gemm_wmma_kernel<128, 32><<<cdiv(Nn, 16), 64, 0, stream>>>(out1, W2, h2, Nn);
  rowdot_kernel<1, 32><<<cdiv(Nn, 256), b256, 0, stream>>>(h2, a_src2, a_dst2, s2, d2, Nn);
  fill_kernel<<<cdiv(Nn, 256), b256, 0, stream>>>(mx2, NEGINF, Nn);
  fill_kernel<<<cdiv(Nn, 256), b256, 0, stream>>>(dn2, 0.f, Nn);
  fill_kernel<<<cdiv(Nn * 32, 256), b256, 0, stream>>>(out2, 0.f, Nn * 32);
  edge_alpha_max_kernel<1><<<cdiv(EE, 256), b256, 0, stream>>>(ei, E, EE, s2, d2, mx2);
  edge_exp_kernel<1><<<cdiv(EE, 256), b256, 0, stream>>>(ei, E, EE, s2, d2, mx2, dn2, ew2);
  inv_kernel<<<cdiv(Nn, 256), b256, 0, stream>>>(dn2, Nn);
  edge_scatter_kernel<32, 1><<<cdiv(EE * 32, 256), b256, 0, stream>>>(ei, E, EE, h2, ew2, dn2, out2);
  bias_sigmoid_kernel<32><<<cdiv(Nn * 32, 256), b256, 0, stream>>>(out2, b2, (float*)d_out, Nn * 32);

  (void)n_in; (void)out_size; (void)ws_size;
}